// FerroelectricBasis_19834158973529
// MI455X (gfx1250) — compile-verified
//
#include <hip/hip_runtime.h>

typedef float v2f __attribute__((ext_vector_type(2)));
typedef float v8f __attribute__((ext_vector_type(8)));

__device__ __forceinline__ float fast_tanh(float y) {
#if __has_builtin(__builtin_amdgcn_tanhf)
    return __builtin_amdgcn_tanhf(y);               // gfx1250 v_tanh_f32
#elif __has_builtin(__builtin_amdgcn_exp2f) && __has_builtin(__builtin_amdgcn_rcpf)
    float e = __builtin_amdgcn_exp2f(-2.885390081777927f * y); // exp(-2y)
    return (1.0f - e) * __builtin_amdgcn_rcpf(1.0f + e);
#else
    return tanhf(y);
#endif
}

// out[b,o] = cb[o] + sum_{i,n} cp*s*tanh(kh*xs - a)
//          = sum_{i,n} [ bias*coef + cp*s*(tanh+1) ]  -  (s @ C1)[b,o]
// where kh=k/2, a=k*Ec/2, cp=Ps*coef, xs=s*x, C1[i,o]=sum_n cp.
// The (s @ C1) term (s in {+-1}, exact in f32) is done with V_WMMA_F32_16X16X4_F32,
// accumulating directly into the elementwise accumulators (already in C/D layout),
// with A = -s so the WMMA performs acc - s@C1.
__global__ __launch_bounds__(512) void FerroelectricBasis_kernel(
    const float* __restrict__ x,   const float* __restrict__ k,
    const float* __restrict__ Ec,  const float* __restrict__ Ps,
    const float* __restrict__ bias,const float* __restrict__ coef,
    float* __restrict__ out)
{
    __shared__ __align__(16) float xs_lds[64][16];   // [i][b_local] = s*x
    __shared__ __align__(16) float s_lds [64][16];   // [i][b_local] = +-1
    __shared__ __align__(16) float c1_lds[64][64];   // [i][o] = sum_n Ps*coef
    __shared__ __align__(16) float oacc  [16][64];   // output tile accumulator

    const int tid   = threadIdx.x;
    const int btile = blockIdx.x * 16;

    // Prologue: stage x tile, branch signs, zero output accumulator.
    for (int idx = tid; idx < 16 * 64; idx += 512) {
        int bl = idx >> 6, i = idx & 63;
        int b  = btile + bl;
        float xv = x[b * 64 + i];
        float pv = (b == 0) ? 0.0f : x[(b - 1) * 64 + i];
        float s  = ((xv - pv) > 0.0f) ? 1.0f : -1.0f;  // sigmoid(5*dx)>0.5 <=> dx>0
        xs_lds[i][bl] = s * xv;
        s_lds [i][bl] = s;
        (&oacc[0][0])[idx] = 0.0f;
    }
    __syncthreads();

    const int wave  = tid >> 5, lane = tid & 31;
    const int otile = (wave & 3) * 16;     // 4 o-tiles of 16
    const int ibase = (wave >> 2) * 16;    // 4 i-splits of 16
    const int m = lane & 15, g = lane >> 4;
    const int o = otile + m;

    const float4* k4 = (const float4*)k;
    const float4* e4 = (const float4*)Ec;
    const float4* p4 = (const float4*)Ps;
    const float4* b4 = (const float4*)bias;
    const float4* c4 = (const float4*)coef;

    v8f   acc   = {0.f,0.f,0.f,0.f,0.f,0.f,0.f,0.f};
    float cbacc = 0.0f;

    for (int i = ibase; i < ibase + 16; ++i) {
        const int p = (i * 64 + o) * 4;          // float4 index, 64B aligned
        float KA[16], EA[16], PA[16], BA[16], CA[16];
        #pragma unroll
        for (int q = 0; q < 4; ++q) {
            ((float4*)KA)[q] = k4[p + q];
            ((float4*)EA)[q] = e4[p + q];
            ((float4*)PA)[q] = p4[p + q];
            ((float4*)BA)[q] = b4[p + q];
            ((float4*)CA)[q] = c4[p + q];
        }
        float xs8[8], s8[8];
        #pragma unroll
        for (int q = 0; q < 2; ++q) {
            ((float4*)xs8)[q] = ((const float4*)&xs_lds[i][g * 8])[q];
            ((float4*)s8 )[q] = ((const float4*)&s_lds [i][g * 8])[q];
        }
        float c1 = 0.0f;
        #pragma unroll
        for (int n = 0; n < 16; ++n) {
            float kh = 0.5f * KA[n];
            float a  = kh * EA[n];
            float cp = PA[n] * CA[n];
            cbacc = fmaf(BA[n], CA[n], cbacc);   // bias*coef constant term
            c1 += cp;
            float na = -a;
            #pragma unroll
            for (int r = 0; r < 8; ++r) {
                float y = fmaf(kh, xs8[r], na);
                float t = fast_tanh(y);
                float u = fmaf(cp, t, cp);       // cp*(t+1)
                acc[r]  = fmaf(s8[r], u, acc[r]);
            }
        }
        c1_lds[i][o] = c1;   // same-wave LDS store; both g-halves write same value
    }

    #pragma unroll
    for (int r = 0; r < 8; ++r) acc[r] += cbacc;

    // acc = (-s_tile) x C1_tile + acc over this wave's 16 i's: 4x K=4 WMMA.
    // A layout (16x4 f32): lanes 0-15 M=0..15 hold K={0,1}; lanes 16-31 hold K={2,3}.
    #pragma unroll
    for (int kk = 0; kk < 4; ++kk) {
        int i0 = ibase + kk * 4 + g * 2;
        v2f A, Bv;
        A.x  = -s_lds[i0][m];
        A.y  = -s_lds[i0 + 1][m];
        Bv.x = c1_lds[i0][o];
        Bv.y = c1_lds[i0 + 1][o];
        acc = __builtin_amdgcn_wmma_f32_16x16x4_f32(
            /*neg_a=*/false, A, /*neg_b=*/false, Bv,
            /*c_mod=*/(short)0, acc, /*reuse_a=*/false, /*reuse_b=*/false);
    }

    // Combine i-split partials: D layout -> b_local = r + 8*g, o = otile + m.
    #pragma unroll
    for (int r = 0; r < 8; ++r)
        atomicAdd(&oacc[r + g * 8][o], acc[r]);
    __syncthreads();

    for (int idx = tid; idx < 16 * 64; idx += 512) {
        int bl = idx >> 6, oo = idx & 63;
        out[(btile + bl) * 64 + oo] = oacc[bl][oo];
    }
}

extern "C" void kernel_launch(void* const* d_in, const int* in_sizes, int n_in,
                              void* d_out, int out_size, void* d_ws, size_t ws_size,
                              hipStream_t stream) {
    const float* x    = (const float*)d_in[0];
    const float* k    = (const float*)d_in[1];
    const float* Ec   = (const float*)d_in[2];
    const float* Ps   = (const float*)d_in[3];
    const float* bias = (const float*)d_in[4];
    const float* coef = (const float*)d_in[5];
    float* out = (float*)d_out;

    const int B = in_sizes[0] / 64;          // 1024
    FerroelectricBasis_kernel<<<B / 16, 512, 0, stream>>>(x, k, Ec, Ps, bias, coef, out);
}